// SpatialAttention_50422916055230
// MI455X (gfx1250) — compile-verified
//
#include <hip/hip_runtime.h>
#include <math.h>
#include <stdint.h>

typedef __attribute__((ext_vector_type(2))) float v2f;
typedef __attribute__((ext_vector_type(8))) float v8f;

#define WMMA_F32(a, b, c) \
  __builtin_amdgcn_wmma_f32_16x16x4_f32(false, (a), false, (b), (short)0, (c), false, false)

constexpr int Bn = 16, Hn = 8, Nn = 2500, Ln = 12, Dn = 128;
constexpr int Tn   = Nn * Ln;     // 30000 flattened (n,l) columns per channel row
constexpr int NT   = 157;         // ceil(N/16) n-tiles
constexpr int QSTR = NT * 192;    // 30144 padded q row stride (keeps edge-tile reads in-bounds)

// Async copy of 16 bytes global -> LDS (tracked by ASYNCcnt).
__device__ __forceinline__ void async_g2l_b128(uint32_t lds_off, const void* gaddr) {
  asm volatile("global_load_async_to_lds_b128 %0, %1, off"
               :: "v"(lds_off), "v"((uint64_t)(uintptr_t)gaddr)
               : "memory");
}
__device__ __forceinline__ void wait_asynccnt0() {
  asm volatile("s_wait_asynccnt 0x0" ::: "memory");
}

// ---------------------------------------------------------------------------
// Kernel 1: fused q/v 1x1 conv + ReLU + per-head softmax on q.
// Grid: B * (Tn/16) blocks, 8 waves; wave h owns output channels [16h,16h+16).
// The shared 128x16 X tile is staged into LDS once via async loads.
// ---------------------------------------------------------------------------
__global__ __launch_bounds__(256) void k_qv(const float* __restrict__ X,
                                            const float* __restrict__ Wq,
                                            const float* __restrict__ bq,
                                            const float* __restrict__ Wv,
                                            const float* __restrict__ bv,
                                            float* __restrict__ qbuf,
                                            float* __restrict__ vbuf) {
  const int blk  = blockIdx.x;
  const int b    = blk / (Tn / 16);
  const int t0   = (blk % (Tn / 16)) * 16;
  const int h    = threadIdx.x >> 5;   // wave id == head
  const int lane = threadIdx.x & 31;
  const int hi   = lane >> 4;
  const int m    = lane & 15;
  const int row  = h * 16 + m;         // A-matrix row (output channel)
  const float* xb = X + (size_t)b * Dn * Tn;

  __shared__ float xt[Dn * 16];        // 8 KB B-operand tile: xt[c*16 + col]
  {
    const uint32_t base = (uint32_t)(uintptr_t)xt;
    // 512 16-byte quarters: idx -> (row = idx>>2, quarter = idx&3)
    for (int j = 0; j < 2; j++) {
      const int idx = threadIdx.x + j * 256;
      const int r = idx >> 2, qt = idx & 3;
      async_g2l_b128(base + (uint32_t)idx * 16,
                     xb + (size_t)r * Tn + t0 + qt * 4);
    }
    wait_asynccnt0();
  }
  __syncthreads();

  v8f accq = {}, accv = {};
  for (int k0 = 0; k0 < Dn; k0 += 4) {
    const int kk = k0 + 2 * hi;
    v2f aq, av, bx;
    aq.x = Wq[row * Dn + kk];  aq.y = Wq[row * Dn + kk + 1];
    av.x = Wv[row * Dn + kk];  av.y = Wv[row * Dn + kk + 1];
    bx.x = xt[kk * 16 + m];
    bx.y = xt[(kk + 1) * 16 + m];
    accq = WMMA_F32(aq, bx, accq);
    accv = WMMA_F32(av, bx, accv);
  }

  // q: relu -> *0.25 -> softmax over the 16 head channels (D rows i+8*hi).
  float s[8];
  float mx = -1e30f;
  for (int i = 0; i < 8; i++) {
    const int c = h * 16 + i + 8 * hi;
    s[i] = fmaxf(accq[i] + bq[c], 0.f) * 0.25f;
    mx = fmaxf(mx, s[i]);
  }
  mx = fmaxf(mx, __shfl_xor(mx, 16, 32));
  float sum = 0.f;
  for (int i = 0; i < 8; i++) { s[i] = __expf(s[i] - mx); sum += s[i]; }
  sum += __shfl_xor(sum, 16, 32);
  const float inv = 1.f / sum;

  for (int i = 0; i < 8; i++) {
    const int c = h * 16 + i + 8 * hi;
    qbuf[(size_t)(b * Dn + c) * QSTR + t0 + m] = s[i] * inv;
    vbuf[(size_t)(b * Dn + c) * Tn   + t0 + m] = fmaxf(accv[i] + bv[c], 0.f);
  }
}

// ---------------------------------------------------------------------------
// Kernel 2a: key = softmax(s_bank * 0.25) over dk (contiguous 16).
// ---------------------------------------------------------------------------
__global__ __launch_bounds__(256) void k_keysm(const float* __restrict__ sb,
                                               float* __restrict__ keysm) {
  const int r = blockIdx.x * blockDim.x + threadIdx.x;
  if (r >= Hn * Ln * Nn) return;
  const float* p = sb + (size_t)r * 16;
  float v[16], mx = -1e30f;
  for (int j = 0; j < 16; j++) { v[j] = p[j] * 0.25f; mx = fmaxf(mx, v[j]); }
  float sum = 0.f;
  for (int j = 0; j < 16; j++) { v[j] = __expf(v[j] - mx); sum += v[j]; }
  const float inv = 1.f / sum;
  float* q = keysm + (size_t)r * 16;
  for (int j = 0; j < 16; j++) q[j] = v[j] * inv;
}

// ---------------------------------------------------------------------------
// Kernel 2: kv[b,h,l] (16x16) = key[h,l]^T (16xN) @ v[b,h,l] (Nx16), K=2500.
// 4 waves split the 625 K=4 chunks; LDS tree reduce.
// ---------------------------------------------------------------------------
__global__ __launch_bounds__(128) void k_kv(const float* __restrict__ keysm,
                                            const float* __restrict__ vbuf,
                                            float* __restrict__ kvbuf) {
  const int blk  = blockIdx.x;          // (b*8+h)*12 + l, l fastest for L2 reuse
  const int l    = blk % Ln;
  const int h    = (blk / Ln) % Hn;
  const int b    = blk / (Ln * Hn);
  const int wave = threadIdx.x >> 5;
  const int lane = threadIdx.x & 31;
  const int hi   = lane >> 4;
  const int m    = lane & 15;           // A: x-row; B: y-column (v channel)
  const float* kp = keysm + (size_t)(h * Ln + l) * Nn * 16;
  const float* vp = vbuf + (size_t)(b * Dn + h * 16 + m) * Tn + l;

  v8f acc = {};
  for (int chunk = wave; chunk < Nn / 4; chunk += 4) {
    const int n0 = chunk * 4 + 2 * hi;
    v2f a, bx;
    a.x  = kp[n0 * 16 + m];        a.y  = kp[(n0 + 1) * 16 + m];
    bx.x = vp[(size_t)n0 * Ln];    bx.y = vp[(size_t)(n0 + 1) * Ln];
    acc = WMMA_F32(a, bx, acc);
  }

  __shared__ float red[4][32][8];
  for (int i = 0; i < 8; i++) red[wave][lane][i] = acc[i];
  __syncthreads();
  if (wave == 0) {
    for (int i = 0; i < 8; i++) {
      const float v = red[0][lane][i] + red[1][lane][i] +
                      red[2][lane][i] + red[3][lane][i];
      kvbuf[(size_t)blk * 256 + (i + 8 * hi) * 16 + m] = v;
    }
  }
}

// ---------------------------------------------------------------------------
// Kernel 3: per (b, 16-n tile): O = blockdiag(kv^T) @ q  (stage 1, LDS),
// then out = relu(Wc @ O + bc) with contiguous 192-float stores (stage 2).
// ---------------------------------------------------------------------------
__global__ __launch_bounds__(256) void k_out(const float* __restrict__ qbuf,
                                             const float* __restrict__ kvbuf,
                                             const float* __restrict__ Wc,
                                             const float* __restrict__ bc,
                                             float* __restrict__ out) {
  const int blk  = blockIdx.x;
  const int b    = blk / NT;
  const int n0   = (blk % NT) * 16;
  const int h    = threadIdx.x >> 5;
  const int lane = threadIdx.x & 31;
  const int hi   = lane >> 4;
  const int m    = lane & 15;
  __shared__ float O[Dn * 208];        // 128 x 192 tile, stride 208 (bank dodge)

  // Stage 1: for each l, O[h*16+y][n_local*12+l] = sum_x kv[x,y]*q[x, col]
  for (int l = 0; l < Ln; l++) {
    const float* kv = kvbuf + (size_t)(((b * Hn) + h) * Ln + l) * 256;
    v8f acc = {};
    for (int k0 = 0; k0 < 16; k0 += 4) {
      const int kk = k0 + 2 * hi;
      v2f a, bx;
      a.x = kv[kk * 16 + m];  a.y = kv[(kk + 1) * 16 + m];   // A = kv^T
      const float* qp = qbuf + (size_t)(b * Dn + h * 16 + kk) * QSTR
                             + (size_t)(n0 + m) * Ln + l;
      bx.x = qp[0];  bx.y = qp[QSTR];
      acc = WMMA_F32(a, bx, acc);
    }
    for (int i = 0; i < 8; i++)
      O[(h * 16 + i + 8 * hi) * 208 + m * 12 + l] = acc[i];
  }
  __syncthreads();

  // Stage 2: rows [16h,16h+16), 12 column sub-tiles of 16 over the 192 cols.
  for (int ct = 0; ct < 12; ct++) {
    v8f acc = {};
    for (int k0 = 0; k0 < Dn; k0 += 4) {
      const int kk = k0 + 2 * hi;
      v2f a, bx;
      a.x  = Wc[(h * 16 + m) * Dn + kk];  a.y = Wc[(h * 16 + m) * Dn + kk + 1];
      bx.x = O[kk * 208 + ct * 16 + m];
      bx.y = O[(kk + 1) * 208 + ct * 16 + m];
      acc = WMMA_F32(a, bx, acc);
    }
    for (int i = 0; i < 8; i++) {
      const int c = h * 16 + i + 8 * hi;
      const int t = n0 * Ln + ct * 16 + m;   // contiguous (n,l) flat index
      if (t < Tn)
        out[(size_t)(b * Dn + c) * Tn + t] = fmaxf(acc[i] + bc[c], 0.f);
    }
  }
}

// ---------------------------------------------------------------------------
extern "C" void kernel_launch(void* const* d_in, const int* in_sizes, int n_in,
                              void* d_out, int out_size, void* d_ws, size_t ws_size,
                              hipStream_t stream) {
  const float* x  = (const float*)d_in[0];
  const float* Wq = (const float*)d_in[1];
  const float* bq = (const float*)d_in[2];
  const float* Wv = (const float*)d_in[3];
  const float* bv = (const float*)d_in[4];
  const float* Wc = (const float*)d_in[5];
  const float* bc = (const float*)d_in[6];
  const float* sb = (const float*)d_in[7];
  float* out = (float*)d_out;

  float* qbuf  = (float*)d_ws;                          // B*D*QSTR   (~247 MB)
  float* vbuf  = qbuf  + (size_t)Bn * Dn * QSTR;        // B*D*Tn     (~246 MB)
  float* keysm = vbuf  + (size_t)Bn * Dn * Tn;          // H*L*N*16   (~15 MB)
  float* kvbuf = keysm + (size_t)Hn * Ln * Nn * 16;     // B*H*L*256  (~1.6 MB)

  hipLaunchKernelGGL(k_keysm, dim3((Hn * Ln * Nn + 255) / 256), dim3(256), 0, stream,
                     sb, keysm);
  hipLaunchKernelGGL(k_qv, dim3(Bn * (Tn / 16)), dim3(256), 0, stream,
                     x, Wq, bq, Wv, bv, qbuf, vbuf);
  hipLaunchKernelGGL(k_kv, dim3(Bn * Hn * Ln), dim3(128), 0, stream,
                     keysm, vbuf, kvbuf);
  hipLaunchKernelGGL(k_out, dim3(Bn * NT), dim3(256), 0, stream,
                     qbuf, kvbuf, Wc, bc, out);
}